// RenderLidarDepth_21775484190829
// MI455X (gfx1250) — compile-verified
//
#include <hip/hip_runtime.h>
#include <stdint.h>

// ---------------- problem constants (from reference) ----------------
#define GXv 256
#define GYv 256
#define GZv 32
#define N_PTS 16384
#define FREE_LBL 17
#define VOXf 0.4f
#define RAY_EPS 1e-9f
#define MIN_DISTf 2.5f
#define MAX_DISTf 144.95682f            // ||(102.4,102.4,6.4)||
#define LXf (-51.2f)
#define LYf (-51.2f)
#define LZf (-3.2f)
#define UXf (51.2f)
#define UYf (51.2f)
#define UZf (3.2f)
#define F_INF __builtin_inff()
#define F_NAN __builtin_nanf("")

// ============================================================================
// Kernel 1: pack occupancy (B,256,256,32) int32 -> per-(b,x,y) 32-bit column
// bitmask (bit z set iff label != FREE).  16 MB -> 512 KB. Streaming read,
// this is essentially the only HBM traffic of the whole pipeline.
// ============================================================================
__global__ __launch_bounds__(256) void
pack_occ_kernel(const int* __restrict__ occ, uint32_t* __restrict__ packed)
{
    const int col = blockIdx.x * 256 + (int)threadIdx.x;   // b*65536 + x*256 + y
    const int4* p = reinterpret_cast<const int4*>(occ) + (size_t)col * 8;
    uint32_t bits = 0u;
#pragma unroll
    for (int k = 0; k < 8; ++k) {
        int4 v = p[k];
        bits |= (v.x != FREE_LBL ? 1u : 0u) << (4 * k + 0);
        bits |= (v.y != FREE_LBL ? 1u : 0u) << (4 * k + 1);
        bits |= (v.z != FREE_LBL ? 1u : 0u) << (4 * k + 2);
        bits |= (v.w != FREE_LBL ? 1u : 0u) << (4 * k + 3);
    }
    packed[col] = bits;
}

// ============================================================================
// Kernel 2: one thread per ray.  Stage the batch's 256 KB column-mask into
// the 320 KB CDNA5 LDS via async global->LDS DMA (ASYNCcnt), then run exact
// Amanatides-Woo DDA with LDS bit tests, then the reference epilogue.
// ============================================================================
__global__ __launch_bounds__(256) void
lidar_ray_kernel(const uint32_t* __restrict__ packed,
                 const float* __restrict__ pts_lidar,
                 const unsigned char* __restrict__ pmask,
                 const float* __restrict__ ego_T,
                 float* __restrict__ out,
                 int nRays)
{
    extern __shared__ uint32_t smem[];              // 65536 dwords = 256 KB

    const int ray = blockIdx.x * 256 + (int)threadIdx.x;
    const int b   = ray >> 14;                      // 16384 rays per batch

    // ---- async global -> LDS staging of this batch's column masks ----
    {
        uint64_t gbase = (uint64_t)(size_t)(packed + (size_t)b * (GXv * GYv));
        uint32_t t = (uint32_t)threadIdx.x;
#pragma unroll 4
        for (int k = 0; k < 64; ++k) {              // 256 thr * 64 * 16B = 256 KB
            uint32_t ldsOff = (uint32_t)(k * 256) * 16u + t * 16u;
            uint64_t gaddr  = gbase + (uint64_t)ldsOff;
            asm volatile("global_load_async_to_lds_b128 %0, %1, off"
                         :: "v"(ldsOff), "v"(gaddr) : "memory");
        }
        asm volatile("s_wait_asynccnt 0" ::: "memory");
    }
    __syncthreads();

    // ---- per-ray setup ----
    const float* Tb = ego_T + b * 16;
    const float r00 = Tb[0],  r01 = Tb[1],  r02 = Tb[2],  tx = Tb[3];
    const float r10 = Tb[4],  r11 = Tb[5],  r12 = Tb[6],  ty = Tb[7];
    const float r20 = Tb[8],  r21 = Tb[9],  r22 = Tb[10], tz = Tb[11];

    const float* p3 = pts_lidar + (size_t)ray * 3;
    const float px = p3[0], py = p3[1], pz = p3[2];
    const bool  msk = pmask[ray] != 0;

    const float sx = tx, sy = ty, sz = tz;                      // start_ego
    const float ex = r00 * px + r01 * py + r02 * pz + tx;       // end_ego
    const float ey = r10 * px + r11 * py + r12 * pz + ty;
    const float ez = r20 * px + r21 * py + r22 * pz + tz;

    const float svx = (sx - LXf) / VOXf, svy = (sy - LYf) / VOXf, svz = (sz - LZf) / VOXf;
    const float evx = (ex - LXf) / VOXf, evy = (ey - LYf) / VOXf, evz = (ez - LZf) / VOXf;
    const float rx = evx - svx, ry = evy - svy, rz = evz - svz;
    const float ray_len = sqrtf(rx * rx + ry * ry + rz * rz);   // voxel units
    const float dx = rx / ray_len, dy = ry / ray_len, dz = rz / ray_len;

    // ---- slab test: forward-ray intersection with the grid ----
    float tEnter = 0.f, tExit = F_INF;
    bool valid = (ray_len > 0.f);
    if (fabsf(dx) > RAY_EPS) {
        float a0 = (0.f - svx) / dx, a1 = (256.f - svx) / dx;
        tEnter = fmaxf(tEnter, fminf(a0, a1));
        tExit  = fminf(tExit,  fmaxf(a0, a1));
    } else if (!(svx >= 0.f && svx < 256.f)) { valid = false; }
    if (fabsf(dy) > RAY_EPS) {
        float a0 = (0.f - svy) / dy, a1 = (256.f - svy) / dy;
        tEnter = fmaxf(tEnter, fminf(a0, a1));
        tExit  = fminf(tExit,  fmaxf(a0, a1));
    } else if (!(svy >= 0.f && svy < 256.f)) { valid = false; }
    if (fabsf(dz) > RAY_EPS) {
        float a0 = (0.f - svz) / dz, a1 = (32.f - svz) / dz;
        tEnter = fmaxf(tEnter, fminf(a0, a1));
        tExit  = fminf(tExit,  fmaxf(a0, a1));
    } else if (!(svz >= 0.f && svz < 32.f)) { valid = false; }

    // ---- DDA traversal ----
    bool  hit = false;
    float first_t = F_INF;
    int   ntrans = 0;

    if (valid && (tEnter < tExit)) {
        float invdx = 0.f, invdy = 0.f, invdz = 0.f;
        int   stpX = 0, stpY = 0, stpZ = 0, offX = 0, offY = 0, offZ = 0;
        float tncx = F_INF, tncy = F_INF, tncz = F_INF;
        if (fabsf(dx) > RAY_EPS) {
            invdx = 1.0f / dx; stpX = dx > 0.f ? 1 : -1; offX = dx > 0.f ? 1 : 0;
            float q = svx + tEnter * dx;
            float plane = (dx > 0.f) ? floorf(q) + 1.0f : floorf(q);
            tncx = (plane - svx) * invdx;
            if (tncx <= tEnter) tncx += fabsf(invdx);
        }
        if (fabsf(dy) > RAY_EPS) {
            invdy = 1.0f / dy; stpY = dy > 0.f ? 1 : -1; offY = dy > 0.f ? 1 : 0;
            float q = svy + tEnter * dy;
            float plane = (dy > 0.f) ? floorf(q) + 1.0f : floorf(q);
            tncy = (plane - svy) * invdy;
            if (tncy <= tEnter) tncy += fabsf(invdy);
        }
        if (fabsf(dz) > RAY_EPS) {
            invdz = 1.0f / dz; stpZ = dz > 0.f ? 1 : -1; offZ = dz > 0.f ? 1 : 0;
            float q = svz + tEnter * dz;
            float plane = (dz > 0.f) ? floorf(q) + 1.0f : floorf(q);
            tncz = (plane - svz) * invdz;
            if (tncz <= tEnter) tncz += fabsf(invdz);
        }
        float tNext = fminf(fminf(tncx, tncy), tncz);
        if (tNext < 1e37f) {
            const float tm = 0.5f * (tEnter + tNext);   // midpoint of first cell
            int ix = (int)floorf(svx + tm * dx);
            int iy = (int)floorf(svy + tm * dy);
            int iz = (int)floorf(svz + tm * dz);
            float tMaxX = stpX ? ((float)(ix + offX) - svx) * invdx : F_INF;
            float tMaxY = stpY ? ((float)(iy + offY) - svy) * invdy : F_INF;
            float tMaxZ = stpZ ? ((float)(iz + offZ) - svz) * invdz : F_INF;
            float tCur = tEnter;
#pragma unroll 1
            for (int it = 0; it < 1200; ++it) {
                if (hit && tCur >= ray_len) break;      // no further contributions
                if ((unsigned)ix >= 256u || (unsigned)iy >= 256u || (unsigned)iz >= 32u)
                    break;                              // convex grid: exited
                uint32_t w = smem[(ix << 8) | iy];      // LDS column bitmask
                if ((w >> iz) & 1u) {
                    if (!hit) { hit = true; first_t = tCur; }
                    if (tCur < ray_len) ntrans++; else break;
                }
                if (tMaxX <= tMaxY && tMaxX <= tMaxZ) {
                    tCur = tMaxX; ix += stpX; tMaxX = ((float)(ix + offX) - svx) * invdx;
                } else if (tMaxY <= tMaxZ) {
                    tCur = tMaxY; iy += stpY; tMaxY = ((float)(iy + offY) - svy) * invdy;
                } else {
                    tCur = tMaxZ; iz += stpZ; tMaxZ = ((float)(iz + offZ) - svz) * invdz;
                }
            }
        }
    }

    // ---- epilogue (matches reference NaN / inf semantics) ----
    const float lidar_depth = sqrtf(px * px + py * py + pz * pz);
    const float depth = hit ? first_t * VOXf : F_INF;

    float rendered, ntransf;
    bool finite_depth;
    if (msk) {
        ntransf = (float)ntrans;
        finite_depth = hit && (lidar_depth > MIN_DISTf);
        float rend = depth < MAX_DISTf ? depth : MAX_DISTf;     // min(depth, MAX)
        float dex = ex - sx, dey = ey - sy, dez = ez - sz;
        float dlen = sqrtf(dex * dex + dey * dey + dez * dez);
        float segx = (dex / dlen) * rend;
        float segy = (dey / dlen) * rend;
        float segz = (dez / dlen) * rend;
        float tax = segx > 0.f ? (UXf - sx) / segx : (segx < 0.f ? (LXf - sx) / segx : F_INF);
        float tay = segy > 0.f ? (UYf - sy) / segy : (segy < 0.f ? (LYf - sy) / segy : F_INF);
        float taz = segz > 0.f ? (UZf - sz) / segz : (segz < 0.f ? (LZf - sz) / segz : F_INF);
        float tcl = fminf(fminf(tax, tay), taz);
        tcl = fminf(fmaxf(tcl, 0.f), 1.f);
        float cx = tcl * segx, cy = tcl * segy, cz = tcl * segz;
        rendered = sqrtf(cx * cx + cy * cy + cz * cz);
    } else {
        ntransf = F_NAN;
        rendered = F_NAN;
        finite_depth = false;
    }
    const bool in_vol = msk && (lidar_depth > MIN_DISTf)
        && ex >= LXf && ex <= UXf && ey >= LYf && ey <= UYf && ez >= LZf && ez <= UZf
        && sx >= LXf && sx <= UXf && sy >= LYf && sy <= UYf && sz >= LZf && sz <= UZf;

    out[0 * nRays + ray] = rendered;
    out[1 * nRays + ray] = ntransf;
    out[2 * nRays + ray] = lidar_depth;
    out[3 * nRays + ray] = in_vol ? 1.0f : 0.0f;
    out[4 * nRays + ray] = finite_depth ? 1.0f : 0.0f;
}

// ============================================================================
extern "C" void kernel_launch(void* const* d_in, const int* in_sizes, int n_in,
                              void* d_out, int out_size, void* d_ws, size_t ws_size,
                              hipStream_t stream)
{
    (void)n_in; (void)out_size; (void)ws_size;
    const int*           occ = (const int*)d_in[0];            // (B,256,256,32) int32
    const float*         pts = (const float*)d_in[1];          // (B,N,3) f32
    const unsigned char* msk = (const unsigned char*)d_in[2];  // (B,N) bool (1 byte)
    const float*         tfm = (const float*)d_in[3];          // (B,4,4) f32
    uint32_t* packed = (uint32_t*)d_ws;                        // (B,256,256) u32, 512 KB
    float*    out    = (float*)d_out;

    const int nRays     = in_sizes[1] / 3;        // B*N = 32768
    const int nCols     = in_sizes[0] / 32;       // B*256*256 = 131072
    const int colBlocks = nCols / 256;            // 512
    const int rayBlocks = nRays / 256;            // 128

    pack_occ_kernel<<<colBlocks, 256, 0, stream>>>(occ, packed);
    lidar_ray_kernel<<<rayBlocks, 256, 256 * 1024, stream>>>(packed, pts, msk, tfm,
                                                             out, nRays);
}